// GraphConvolution_43559558316209
// MI455X (gfx1250) — compile-verified
//
#include <hip/hip_runtime.h>
#include <hip/hip_bf16.h>

// GCN layer: out = spmm(adj, x) @ W + bias
// Reordered as:   z = x @ W  (dense WMMA GEMM)
//                 out = bias (broadcast init)
//                 out[n, r, :] += v * z[n, c, :]  for each edge (r, c, v)
// Valid because (A X) W == A (X W).

typedef __attribute__((ext_vector_type(2))) float v2f;
typedef __attribute__((ext_vector_type(8))) float v8f;

#define GCN_N 8
#define GCN_M 50000
#define GCN_FIN 32
#define GCN_FOUT 32
#define GCN_E 800000

// ---------------------------------------------------------------------------
// Kernel 1: z[R, 32] = x[R, 32] @ W[32, 32], R = N*M = 400000 flattened rows.
// One wave computes a 16-row x 32-col tile using V_WMMA_F32_16X16X4_F32,
// chaining K=32 as 8 steps of K=4, with two accumulators for the two 16-col
// panels. 256-thread blocks = 8 waves = 8 row-tiles per block.
// A layout (32-bit 16x4): lane L holds row M=L%16; VGPR0 = K = 2*(L/16),
// VGPR1 = K = 2*(L/16)+1. B (4x16) mirrors: lane holds col N=L%16, same K pair.
// C/D (16x16 f32): VGPR j = row j (lanes 0-15) / row j+8 (lanes 16-31), N=L%16.
// ---------------------------------------------------------------------------
__global__ __launch_bounds__(256) void gcn_gemm_xw(const float* __restrict__ x,
                                                   const float* __restrict__ W,
                                                   float* __restrict__ z) {
    const int lane = threadIdx.x & 31;
    const int waveInBlock = threadIdx.x >> 5;
    const long tile = (long)blockIdx.x * 8 + waveInBlock;   // 16-row tile index
    const long rowBase = tile * 16;

    const int m  = lane & 15;          // row within tile (A) / col within panel (B,C)
    const int kq = (lane >> 4) << 1;   // 0 or 2: which K pair this lane holds

    const float* __restrict__ xr = x + (rowBase + m) * GCN_FIN;

    v8f c0 = {};   // columns 0..15
    v8f c1 = {};   // columns 16..31

    #pragma unroll
    for (int kb = 0; kb < GCN_FIN; kb += 4) {
        // A fragment: two consecutive K values for this lane's row.
        v2f a = *(const v2f*)(xr + kb + kq);   // 8-byte aligned (kb+kq even)

        // B fragments: W[k][n] for the same K pair, both column panels.
        v2f b0, b1;
        b0.x = W[(kb + kq)     * GCN_FOUT + m];
        b0.y = W[(kb + kq + 1) * GCN_FOUT + m];
        b1.x = W[(kb + kq)     * GCN_FOUT + 16 + m];
        b1.y = W[(kb + kq + 1) * GCN_FOUT + 16 + m];

        c0 = __builtin_amdgcn_wmma_f32_16x16x4_f32(false, a, false, b0,
                                                   (short)0, c0, false, false);
        c1 = __builtin_amdgcn_wmma_f32_16x16x4_f32(false, a, false, b1,
                                                   (short)0, c1, false, false);
    }

    // Store: VGPR j -> row (j + 8*(lane>=16)), col = lane%16 (+16 for panel 1).
    const int rhalf = lane >> 4;
    #pragma unroll
    for (int j = 0; j < 8; ++j) {
        const long r = rowBase + j + 8 * rhalf;
        z[r * GCN_FOUT + m]        = c0[j];
        z[r * GCN_FOUT + 16 + m]   = c1[j];
    }
}

// ---------------------------------------------------------------------------
// Kernel 2: out[n, m, :] = bias[:]  (float4 vectorized broadcast)
// Must run every call: harness poisons d_out once and never re-poisons, and
// zero-in-degree rows still need the bias term.
// ---------------------------------------------------------------------------
__global__ __launch_bounds__(256) void gcn_init_bias(float* __restrict__ out,
                                                     const float* __restrict__ bias,
                                                     int total4) {
    int i = blockIdx.x * blockDim.x + threadIdx.x;
    if (i < total4) {
        int o4 = i & (GCN_FOUT / 4 - 1);   // which float4 of the 32-wide bias
        float4 b = ((const float4*)bias)[o4];
        ((float4*)out)[i] = b;
    }
}

// ---------------------------------------------------------------------------
// Kernel 3: edge scatter. One block per edge; wave w handles batch n=w,
// lane handles feature f. Edge metadata is block-uniform -> scalar loads.
// Per-wave gathers/atomics hit 128B-contiguous spans; the whole z/out working
// set (2 x 51 MB) is L2-resident on MI455X (192 MB L2).
// ---------------------------------------------------------------------------
__global__ __launch_bounds__(256) void gcn_scatter(const float* __restrict__ z,
                                                   const int* __restrict__ adj_rows,
                                                   const int* __restrict__ adj_cols,
                                                   const float* __restrict__ adj_vals,
                                                   float* __restrict__ out) {
    const int e = blockIdx.x;
    const int n = threadIdx.x >> 5;
    const int f = threadIdx.x & 31;

    const int r = adj_rows[e];
    const int c = adj_cols[e];
    const float v = adj_vals[e];

    const float val = v * z[((long)n * GCN_M + c) * GCN_FOUT + f];
    unsafeAtomicAdd(&out[((long)n * GCN_M + r) * GCN_FOUT + f], val);
}

extern "C" void kernel_launch(void* const* d_in, const int* in_sizes, int n_in,
                              void* d_out, int out_size, void* d_ws, size_t ws_size,
                              hipStream_t stream) {
    const float* x    = (const float*)d_in[0];
    const int*   ar   = (const int*)d_in[1];
    const int*   ac   = (const int*)d_in[2];
    const float* av   = (const float*)d_in[3];
    const float* W    = (const float*)d_in[4];
    const float* bias = (const float*)d_in[5];
    float* out = (float*)d_out;
    float* z   = (float*)d_ws;   // N*M*FOUT floats = 51.2 MB scratch

    const int rows = GCN_N * GCN_M;          // 400000, divisible by 16*8
    const int gemmBlocks = rows / 16 / 8;    // 3125 blocks of 8 waves

    gcn_gemm_xw<<<gemmBlocks, 256, 0, stream>>>(x, W, z);

    const int total4 = rows * GCN_FOUT / 4;  // 3.2M float4 stores
    gcn_init_bias<<<(total4 + 255) / 256, 256, 0, stream>>>(out, bias, total4);

    gcn_scatter<<<GCN_E, 256, 0, stream>>>(z, ar, ac, av, out);
}